// ClassifyNet_53919019434673
// MI455X (gfx1250) — compile-verified
//
#include <hip/hip_runtime.h>
#include <hip/hip_bf16.h>

// ---------------------------------------------------------------------------
// Problem constants (from reference)
// ---------------------------------------------------------------------------
#define LAYERS_TOT 13
#define BATCH      256
#define SEQ        128
#define HID        768
#define FEATD      3072            // 4 * 768
#define CDIM       8192
#define D1DIM      3000
#define GRP        16
#define TOPK       10
#define NCAND      160             // TOPK * GRP

typedef __attribute__((ext_vector_type(2))) float v2f;
typedef __attribute__((ext_vector_type(8))) float v8f;

// ---------------------------------------------------------------------------
// Kernel A: feat[b, i*768 + h] = hidden_states[12-i, b, 0, h]
// ---------------------------------------------------------------------------
__global__ void build_feat_kernel(const float* __restrict__ hs,
                                  float* __restrict__ feat) {
    const int b = blockIdx.x;
    for (int idx = threadIdx.x; idx < FEATD; idx += blockDim.x) {
        const int blk   = idx / HID;                  // 0..3
        const int h     = idx - blk * HID;
        const int layer = (LAYERS_TOT - 1) - blk;     // 12, 11, 10, 9
        feat[(size_t)b * FEATD + idx] =
            hs[((size_t)layer * BATCH + b) * (size_t)(SEQ * HID) + h];
    }
}

// ---------------------------------------------------------------------------
// Kernels B/D: fp32 WMMA GEMM  C[M,N] = A[M,K] * B[K,N] + bias[N]
// Compile-time N/K/M -> constant-stride pointer increments, immediate offsets.
// One wave computes a 64x16 strip (4 M-tiles reuse each B fragment).
// Edge N-tile handled by clamped loads + cndmask (no exec divergence).
// 1-stage software pipeline: fragments for k+4 load while WMMAs for k run.
// V_WMMA_F32_16X16X4_F32 lane layouts (ISA 7.12.2):
//   A 16x4 : lanes 0-15 hold K=k0,k0+1 (VGPR0,1); lanes 16-31 K=k0+2,k0+3
//   B 4x16 : col = lane&15; lanes 0-15 rows k0/k0+1; lanes 16-31 k0+2/k0+3
//   C 16x16: VGPR i -> M = i + 8*(lane>=16), N = lane&15
// ---------------------------------------------------------------------------
template <int N, int K, int M>
__global__ void gemm_wmma_f32_kernel(const float* __restrict__ A,
                                     const float* __restrict__ Bm,
                                     const float* __restrict__ bias,
                                     float* __restrict__ C) {
    constexpr int  NTILES = (N + 15) / 16;
    constexpr bool FULL   = (N % 16) == 0;
    constexpr int  AS16   = 16 * K;   // A stride between M-tiles (elements)
    constexpr int  BS4    = 4 * N;    // B advance per k-step (elements)

    const int wave   = (blockIdx.x * blockDim.x + threadIdx.x) >> 5;
    const int lane   = threadIdx.x & 31;
    const int ntile  = wave % NTILES;
    const int mblock = wave / NTILES;
    const int m0     = mblock * 64;
    if (m0 >= M) return;

    const int  n0    = ntile * 16;
    const int  col   = lane & 15;
    const int  kpair = (lane >> 4) << 1;          // 0 or 2
    const int  colg  = n0 + col;
    const bool colok = FULL || (colg < N);
    const int  colc  = FULL ? colg : (colok ? colg : (N - 1));  // in-bounds

    const float* ap = A  + (size_t)(m0 + (lane & 15)) * K + kpair;
    const float* bp = Bm + (size_t)kpair * N + colc;

    v8f acc0 = {}, acc1 = {}, acc2 = {}, acc3 = {};

    // ---- prologue: fragments for k0 = 0 ----
    v2f b;
    b[0] = bp[0];
    b[1] = bp[N];
    if (!FULL && !colok) { b[0] = 0.0f; b[1] = 0.0f; }
    v2f a0 = *(const v2f*)(ap);
    v2f a1 = *(const v2f*)(ap + AS16);
    v2f a2 = *(const v2f*)(ap + 2 * AS16);
    v2f a3 = *(const v2f*)(ap + 3 * AS16);

    for (int k0 = 0; k0 < K - 4; k0 += 4) {
        const float* bpn = bp + BS4;
        const float* apn = ap + 4;
        __builtin_prefetch(bpn + 8 * BS4, 0, 1);   // stream weights ahead

        // loads for k0+4 issue before the WMMAs for k0
        v2f bn;
        bn[0] = bpn[0];
        bn[1] = bpn[N];
        if (!FULL && !colok) { bn[0] = 0.0f; bn[1] = 0.0f; }
        v2f an0 = *(const v2f*)(apn);
        v2f an1 = *(const v2f*)(apn + AS16);
        v2f an2 = *(const v2f*)(apn + 2 * AS16);
        v2f an3 = *(const v2f*)(apn + 3 * AS16);

        acc0 = __builtin_amdgcn_wmma_f32_16x16x4_f32(false, a0, false, b,
                                                     (short)0, acc0, false, false);
        acc1 = __builtin_amdgcn_wmma_f32_16x16x4_f32(false, a1, false, b,
                                                     (short)0, acc1, false, false);
        acc2 = __builtin_amdgcn_wmma_f32_16x16x4_f32(false, a2, false, b,
                                                     (short)0, acc2, false, false);
        acc3 = __builtin_amdgcn_wmma_f32_16x16x4_f32(false, a3, false, b,
                                                     (short)0, acc3, false, false);

        a0 = an0; a1 = an1; a2 = an2; a3 = an3; b = bn;
        ap = apn; bp = bpn;
    }

    // ---- epilogue: last k-step ----
    acc0 = __builtin_amdgcn_wmma_f32_16x16x4_f32(false, a0, false, b,
                                                 (short)0, acc0, false, false);
    acc1 = __builtin_amdgcn_wmma_f32_16x16x4_f32(false, a1, false, b,
                                                 (short)0, acc1, false, false);
    acc2 = __builtin_amdgcn_wmma_f32_16x16x4_f32(false, a2, false, b,
                                                 (short)0, acc2, false, false);
    acc3 = __builtin_amdgcn_wmma_f32_16x16x4_f32(false, a3, false, b,
                                                 (short)0, acc3, false, false);

    if (!colok) return;
    const float bb    = bias[colg];
    const int   rbase = (lane >> 4) * 8;
#pragma unroll
    for (int i = 0; i < 8; ++i) {
        const int r = rbase + i;
        C[(size_t)(m0 + r)      * N + colg] = acc0[i] + bb;
        C[(size_t)(m0 + 16 + r) * N + colg] = acc1[i] + bb;
        C[(size_t)(m0 + 32 + r) * N + colg] = acc2[i] + bb;
        C[(size_t)(m0 + 48 + r) * N + colg] = acc3[i] + bb;
    }
}

// ---------------------------------------------------------------------------
// Kernel C: top-10 indices per row (JAX tie-break: lowest index first)
// ---------------------------------------------------------------------------
__global__ void topk10_kernel(const float* __restrict__ logits,
                              int* __restrict__ outidx) {
    __shared__ float vals[CDIM];        // 32 KB of the 320 KB WGP LDS
    __shared__ float rv[256];
    __shared__ int   ri[256];

    const int b   = blockIdx.x;
    const int tid = threadIdx.x;
    const float* row = logits + (size_t)b * CDIM;

    for (int j = tid; j < CDIM; j += 256) vals[j] = row[j];
    __syncthreads();

    for (int t = 0; t < TOPK; ++t) {
        float best = -3.402823466e38f;
        int   bi   = 0x7fffffff;
        for (int j = tid; j < CDIM; j += 256) {
            const float v = vals[j];
            if (v > best) { best = v; bi = j; }
        }
        rv[tid] = best;
        ri[tid] = bi;
        __syncthreads();
        for (int s = 128; s > 0; s >>= 1) {
            if (tid < s) {
                const float ov = rv[tid + s];
                const int   oi = ri[tid + s];
                if (ov > rv[tid] || (ov == rv[tid] && oi < ri[tid])) {
                    rv[tid] = ov; ri[tid] = oi;
                }
            }
            __syncthreads();
        }
        if (tid == 0) {
            outidx[b * TOPK + t] = ri[0];
            vals[ri[0]] = -3.402823466e38f;
        }
        __syncthreads();
    }
}

// ---------------------------------------------------------------------------
// Kernel E: out[b,c] = dot(embed_table[group_y[idx[b,c/16], c%16]], emb[b])
// emb[b] staged in LDS (float4); 8 waves stream candidate rows; wave32 reduce.
// ---------------------------------------------------------------------------
__global__ void gather_dot_kernel(const float* __restrict__ emb,
                                  const int* __restrict__ topidx,
                                  const int* __restrict__ group_y,
                                  const float* __restrict__ etab,
                                  float* __restrict__ out) {
    __shared__ float4 semb[D1DIM / 4];   // 750 float4 = 12 KB
    __shared__ int    cand[NCAND];

    const int b   = blockIdx.x;
    const int tid = threadIdx.x;

    const float4* erow = (const float4*)(emb + (size_t)b * D1DIM);
    for (int j = tid; j < D1DIM / 4; j += 256) semb[j] = erow[j];
    if (tid < NCAND) {
        const int ti = topidx[b * TOPK + (tid >> 4)];
        cand[tid] = group_y[(size_t)ti * GRP + (tid & 15)];
    }
    __syncthreads();

    const int wid  = tid >> 5;
    const int lane = tid & 31;
    for (int c = wid; c < NCAND; c += 8) {
        const float4* grow = (const float4*)(etab + (size_t)cand[c] * D1DIM);
        float s = 0.0f;
        for (int j = lane; j < D1DIM / 4; j += 32) {   // 512 B / wave-step
            const float4 e = semb[j];
            const float4 g = grow[j];
            s += e.x * g.x + e.y * g.y + e.z * g.z + e.w * g.w;
        }
#pragma unroll
        for (int off = 16; off > 0; off >>= 1) s += __shfl_down(s, off, 32);
        if (lane == 0) out[b * NCAND + c] = s;
    }
}

// ---------------------------------------------------------------------------
// Launch
// ---------------------------------------------------------------------------
extern "C" void kernel_launch(void* const* d_in, const int* in_sizes, int n_in,
                              void* d_out, int out_size, void* d_ws, size_t ws_size,
                              hipStream_t stream) {
    const float* hs    = (const float*)d_in[0];   // (13,256,128,768) f32
    // d_in[1] = labels (unused by the reference output)
    const float* W1    = (const float*)d_in[2];   // (3072, 8192)
    const float* b1    = (const float*)d_in[3];   // (8192,)
    const float* W2    = (const float*)d_in[4];   // (8192, 3000)
    const float* b2    = (const float*)d_in[5];   // (3000,)
    const float* etab  = (const float*)d_in[6];   // (65536, 3000)
    const int*   gy    = (const int*)  d_in[7];   // (8192, 16)
    float*       out   = (float*)d_out;           // (256, 160)

    // Workspace layout (~14 MB)
    float* feat   = (float*)d_ws;                              // 256*3072
    float* logits = feat + (size_t)BATCH * FEATD;              // 256*8192
    int*   tidx   = (int*)(logits + (size_t)BATCH * CDIM);     // 256*10 (pad 2560)
    float* emb    = (float*)(tidx + 2560);                     // 256*3000

    // 1. feat gather
    build_feat_kernel<<<BATCH, 256, 0, stream>>>(hs, feat);

    // 2. logits = feat @ W1 + b1 : M=256, N=8192, K=3072
    //    waves = (256/64) * (8192/16) = 2048 -> 256 blocks of 8 waves
    gemm_wmma_f32_kernel<CDIM, FEATD, BATCH>
        <<<256, 256, 0, stream>>>(feat, W1, b1, logits);

    // 3. top-10 per row
    topk10_kernel<<<BATCH, 256, 0, stream>>>(logits, tidx);

    // 4. emb = logits @ W2 + b2 : M=256, N=3000, K=8192
    //    ntiles = ceil(3000/16) = 188; waves = 4*188 = 752 -> 94 blocks
    gemm_wmma_f32_kernel<D1DIM, CDIM, BATCH>
        <<<94, 256, 0, stream>>>(logits, W2, b2, emb);

    // 5. candidate gather + batched dot products
    gather_dot_kernel<<<BATCH, 256, 0, stream>>>(emb, tidx, gy, etab, out);
}